// jacobinNet_644245094832
// MI455X (gfx1250) — compile-verified
//
#include <hip/hip_runtime.h>

#define B_SZ  16
#define D_IN  512
#define D_H   2048
#define D_OUT 512

typedef __attribute__((ext_vector_type(2))) float v2f;
typedef __attribute__((ext_vector_type(4))) float v4f;
typedef __attribute__((ext_vector_type(8))) float v8f;

// ---------------------------------------------------------------------------
// Kernel 1: zero-fill the whole [16, 512, 16, 512] Jacobian (block-diagonal;
// off-diagonal batch blocks are exactly zero and must be written).
// Non-temporal: 256 MB stream, keep it out of L2.
// ---------------------------------------------------------------------------
__global__ void jac_zero_kernel(v4f* __restrict__ out, long n4) {
  long i = (long)blockIdx.x * blockDim.x + threadIdx.x;
  long stride = (long)gridDim.x * blockDim.x;
  v4f z = {};
  for (; i < n4; i += stride) __builtin_nontemporal_store(z, &out[i]);
}

// ---------------------------------------------------------------------------
// Kernel 2: relu-derivative mask[b][h] = (x[b]·W1[:,h] + b1[h] > 0) ? 1 : 0
// Consecutive threads -> consecutive h -> coalesced W1 reads.
// ---------------------------------------------------------------------------
__global__ void jac_mask_kernel(const float* __restrict__ x,
                                const float* __restrict__ W1,
                                const float* __restrict__ b1,
                                float* __restrict__ mask) {
  int idx = blockIdx.x * blockDim.x + threadIdx.x;   // b*D_H + h
  int b = idx / D_H;
  int h = idx - b * D_H;
  float acc = b1[h];
  const float* xr = x + (size_t)b * D_IN;
  for (int i = 0; i < D_IN; ++i)
    acc = fmaf(xr[i], W1[(size_t)i * D_H + h], acc);
  mask[idx] = (acc > 0.f) ? 1.f : 0.f;
}

// ---------------------------------------------------------------------------
// Kernel 3: per-batch GEMM  J_b[o,i] = sum_h (W2[h,o]*mask[b,h]) * W1[i,h]
// M=o (512), N=i (512), K=h (2048).  Block = 128 threads = 4 waves.
// Block tile = 128(M) x 64(N); wave w owns M-rows [32w, 32w+32) -> 8 x v8f
// accumulators (2 M-subtiles x 4 N-subtiles).  K staged in chunks of 32.
// LDS: A[o][k] stride 33 (odd -> conflict-free transpose stores and frag
// reads), B[i][k] stride 36 (float4-aligned stores, aligned b64 frag reads).
// Each lane's K-pair is contiguous -> ds_load_b64 lands in adjacent VGPRs.
// ---------------------------------------------------------------------------
__global__ __launch_bounds__(128)
void jac_gemm_kernel(const float* __restrict__ W1,
                     const float* __restrict__ W2,
                     const float* __restrict__ mask,
                     float* __restrict__ out) {
  __shared__ float Asl[128 * 33];   // A_lds[o][k]  (16.9 KB)
  __shared__ float Bsl[64 * 36];    // B_lds[i][k]  ( 9.2 KB)

  const int tileId = blockIdx.x;         // 0..31  (4 o-tiles x 8 i-tiles)
  const int b      = blockIdx.y;         // 0..15
  const int o0 = (tileId >> 3) * 128;
  const int i0 = (tileId & 7) * 64;

  const int tid  = threadIdx.x;          // 0..127
  const int wave = tid >> 5;             // 0..3
  const int lane = tid & 31;
  const int l16  = lane & 15;
  const int hsel = lane >> 4;            // half-wave: K pairs {0,1} vs {2,3}
  const int m0   = wave * 32;            // 32 M-rows per wave

  v8f acc[8];                            // [mt][nt] -> mt*4 + nt
#pragma unroll
  for (int t = 0; t < 8; ++t) acc[t] = (v8f){};

  const float* maskb = mask + (size_t)b * D_H;

  for (int h0 = 0; h0 < D_H; h0 += 32) {
    // prefetch next K-chunk of W2 (speculative, NT path)
    const int hp = (h0 + 32 < D_H) ? (h0 + 32) : h0;
    __builtin_prefetch(&W2[(size_t)(hp + (tid >> 2)) * D_OUT + o0 + ((tid & 3) << 5)], 0, 0);

    // ---- stage A tile (transpose): Asl[o][k] = W2[h0+k][o0+o] * mask[h0+k]
    //      4096 elems / 128 thr = 32 each; global reads 128-wide contig in o.
#pragma unroll
    for (int r = 0; r < 32; ++r) {
      int e = r * 128 + tid;
      int k = e >> 7;                     // 0..31
      int o = e & 127;                    // 0..127
      Asl[o * 33 + k] =
          W2[(size_t)(h0 + k) * D_OUT + (o0 + o)] * maskb[h0 + k];
    }
    // ---- stage B tile (straight copy): Bsl[i][k] = W1[i0+i][h0+k]
    //      512 float4 / 128 thr = 4 each; 16B-aligned both sides.
#pragma unroll
    for (int r = 0; r < 4; ++r) {
      int e = r * 128 + tid;              // float4 index
      int i = e >> 3;                     // 0..63
      int kq = e & 7;                     // float4 slot within row
      v4f v = *(const v4f*)&W1[(size_t)(i0 + i) * D_H + h0 + (kq << 2)];
      *(v4f*)&Bsl[i * 36 + (kq << 2)] = v;
    }
    __syncthreads();

    // ---- 8 K-steps of 4: 2 A-frags + 4 B-frags -> 8 WMMA per wave per step
#pragma unroll
    for (int kk = 0; kk < 8; ++kk) {
      const int kb = kk * 4 + hsel * 2;   // this half-wave's K pair (even)
      v2f a0, a1;
      a0.x = Asl[(m0 +      l16) * 33 + kb];
      a0.y = Asl[(m0 +      l16) * 33 + kb + 1];
      a1.x = Asl[(m0 + 16 + l16) * 33 + kb];
      a1.y = Asl[(m0 + 16 + l16) * 33 + kb + 1];
#pragma unroll
      for (int nt = 0; nt < 4; ++nt) {
        v2f bf = *(const v2f*)&Bsl[(nt * 16 + l16) * 36 + kb];  // 8B aligned
        acc[nt] = __builtin_amdgcn_wmma_f32_16x16x4_f32(
            false, a0, false, bf, (short)0, acc[nt], false, false);
        acc[4 + nt] = __builtin_amdgcn_wmma_f32_16x16x4_f32(
            false, a1, false, bf, (short)0, acc[4 + nt], false, false);
      }
    }
    __syncthreads();
  }

  // ---- epilogue: write diag block  out[b, o, b, i]   (non-temporal)
  // D layout: VGPR v, lanes 0-15 -> M=v, lanes 16-31 -> M=v+8; N = lane&15.
  const size_t outRow = (size_t)B_SZ * D_IN;                 // 8192
  float* outb = out + (size_t)b * D_OUT * outRow + (size_t)b * D_IN;
#pragma unroll
  for (int mt = 0; mt < 2; ++mt) {
#pragma unroll
    for (int nt = 0; nt < 4; ++nt) {
      const int i = i0 + nt * 16 + l16;
#pragma unroll
      for (int vg = 0; vg < 8; ++vg) {
        const int o = o0 + m0 + mt * 16 + vg + hsel * 8;
        __builtin_nontemporal_store(acc[mt * 4 + nt][vg],
                                    &outb[(size_t)o * outRow + i]);
      }
    }
  }
}

// ---------------------------------------------------------------------------
extern "C" void kernel_launch(void* const* d_in, const int* in_sizes, int n_in,
                              void* d_out, int out_size, void* d_ws, size_t ws_size,
                              hipStream_t stream) {
  const float* x  = (const float*)d_in[0];   // [16, 512]
  const float* W1 = (const float*)d_in[1];   // [512, 2048]
  const float* b1 = (const float*)d_in[2];   // [2048]
  const float* W2 = (const float*)d_in[3];   // [2048, 512]
  // d_in[4] = b2 (drops out of the Jacobian); d_in[5], d_in[6] = flags, ignored.

  float* out  = (float*)d_out;               // 16*512*16*512 = 67,108,864 f32
  float* mask = (float*)d_ws;                // 16*2048 f32 scratch

  // 1) zero-fill full output (off-diagonal batch blocks are exactly zero)
  long n4 = (long)out_size / 4;
  jac_zero_kernel<<<2048, 256, 0, stream>>>((v4f*)out, n4);

  // 2) relu mask
  jac_mask_kernel<<<(B_SZ * D_H) / 256, 256, 0, stream>>>(x, W1, b1, mask);

  // 3) per-batch Jacobian GEMM: grid = (32 tiles, 16 batches), 128 threads
  dim3 grid(32, B_SZ);
  jac_gemm_kernel<<<grid, 128, 0, stream>>>(W1, W2, mask, out);
}